// MultiHashGrid_86208583566130
// MI455X (gfx1250) — compile-verified
//
#include <hip/hip_runtime.h>

// Instant-NGP multi-hash-grid forward, tuned for MI455X (gfx1250, wave32).
//
// Work layout: one thread per (point, level) pair. g = tid & 15 (level),
// b = tid >> 4 (point). Output is [B, 16 levels, 2 feats] f32, so lane-
// consecutive threads write consecutive float2 -> fully coalesced 256B/wave
// nontemporal stores (keeps the 67MB of hash tables resident in the 192MB L2,
// which is the entire performance game for this gather-bound kernel).

typedef float v2f __attribute__((ext_vector_type(2)));

#define HG_LOG2_T 19
#define HG_TABLE_SIZE (1u << HG_LOG2_T)
#define HG_MASK (HG_TABLE_SIZE - 1u)
#define HG_P1 2654435761u
#define HG_P2 805459861u

// floor(16 * (2048/16)^(i/15)) for i = 0..15, matching the reference's
// float64 resolution schedule cast to f32.
__device__ __constant__ float kRes[16] = {
    16.f,  22.f,  30.f,  42.f,  58.f,   80.f,   111.f,  153.f,
    212.f, 294.f, 406.f, 561.f, 776.f, 1072.f, 1482.f, 2048.f};

__global__ __launch_bounds__(256) void hashgrid_fwd_kernel(
    const float* __restrict__ x,       // [B,3]
    const float* __restrict__ tables,  // [16, 2^19, 2]
    v2f* __restrict__ out,             // [B,16] of float2
    int npoints) {
  const int tid = blockIdx.x * blockDim.x + threadIdx.x;
  const int g = tid & 15;
  const int b = tid >> 4;
  if (b >= npoints) return;

  // x is streamed once; NT hint so it doesn't displace table lines in L2.
  // 16 lanes share a point -> the identical addresses coalesce to one request.
  const float px = __builtin_nontemporal_load(x + 3 * b + 0);
  const float py = __builtin_nontemporal_load(x + 3 * b + 1);
  const float pz = __builtin_nontemporal_load(x + 3 * b + 2);

  v2f acc = {0.f, 0.f};

  const bool inbox = (px >= -1.f) && (px <= 1.f) && (py >= -1.f) &&
                     (py <= 1.f) && (pz >= -1.f) && (pz <= 1.f);
  if (inbox) {  // ~32% of N(0,1) points; skip gathers for the rest
    const float r = kRes[g];
    // Match reference order: ((x + 1) * 0.5) * r
    const float xn = (px + 1.f) * 0.5f * r;
    const float yn = (py + 1.f) * 0.5f * r;
    const float zn = (pz + 1.f) * 0.5f * r;
    const float fx = floorf(xn), fy = floorf(yn), fz = floorf(zn);
    const float wx = xn - fx, wy = yn - fy, wz = zn - fz;

    // (u+1)*P == u*P + P mod 2^32, so 3 mads give all 8 corner hash terms.
    const unsigned hx0 = (unsigned)(int)fx;
    const unsigned hx1 = hx0 + 1u;
    const unsigned hy0 = (unsigned)(int)fy * HG_P1;
    const unsigned hy1 = hy0 + HG_P1;
    const unsigned hz0 = (unsigned)(int)fz * HG_P2;
    const unsigned hz1 = hz0 + HG_P2;

    const v2f* __restrict__ tbl =
        (const v2f*)tables + (size_t)g * HG_TABLE_SIZE;

    // Issue all 8 gathers up front for MLP; default temporal hint keeps the
    // tables hot in L2 (fully resident: 67MB of 192MB).
    const v2f f000 = tbl[(hx0 ^ hy0 ^ hz0) & HG_MASK];
    const v2f f001 = tbl[(hx0 ^ hy0 ^ hz1) & HG_MASK];
    const v2f f010 = tbl[(hx0 ^ hy1 ^ hz0) & HG_MASK];
    const v2f f011 = tbl[(hx0 ^ hy1 ^ hz1) & HG_MASK];
    const v2f f100 = tbl[(hx1 ^ hy0 ^ hz0) & HG_MASK];
    const v2f f101 = tbl[(hx1 ^ hy0 ^ hz1) & HG_MASK];
    const v2f f110 = tbl[(hx1 ^ hy1 ^ hz0) & HG_MASK];
    const v2f f111 = tbl[(hx1 ^ hy1 ^ hz1) & HG_MASK];

    // Trilerp weights, corner (i,j,k) weight = ((wi * wj) * wk) as in ref.
    const float wx0 = 1.f - wx, wy0 = 1.f - wy, wz0 = 1.f - wz;
    const float wxy00 = wx0 * wy0, wxy01 = wx0 * wy;
    const float wxy10 = wx * wy0, wxy11 = wx * wy;
    const float w000 = wxy00 * wz0, w001 = wxy00 * wz;
    const float w010 = wxy01 * wz0, w011 = wxy01 * wz;
    const float w100 = wxy10 * wz0, w101 = wxy10 * wz;
    const float w110 = wxy11 * wz0, w111 = wxy11 * wz;

    acc = f000 * w000;
    acc += f001 * w001;
    acc += f010 * w010;
    acc += f011 * w011;
    acc += f100 * w100;
    acc += f101 * w101;
    acc += f110 * w110;
    acc += f111 * w111;
  }

  // Write-once output: NT store, coalesced 8B/lane.
  __builtin_nontemporal_store(acc, out + tid);
}

extern "C" void kernel_launch(void* const* d_in, const int* in_sizes, int n_in,
                              void* d_out, int out_size, void* d_ws,
                              size_t ws_size, hipStream_t stream) {
  const float* x = (const float*)d_in[0];       // [B,3] f32
  const float* tables = (const float*)d_in[1];  // [16, 2^19, 2] f32
  v2f* out = (v2f*)d_out;                       // [B, 32] f32 viewed as [B,16] v2f

  const int npoints = in_sizes[0] / 3;
  const long long total = (long long)npoints * 16;
  const int block = 256;  // 8 wave32s per workgroup
  const int grid = (int)((total + block - 1) / block);

  hashgrid_fwd_kernel<<<grid, block, 0, stream>>>(x, tables, out, npoints);
}